// MetScore_16423954940083
// MI455X (gfx1250) — compile-verified
//
#include <hip/hip_runtime.h>

// ---------------------------------------------------------------------------
// MetScore for MI455X (gfx1250, wave32)
//
// 442 MB read @ 23.3 TB/s => ~19 us floor. Single streaming pass (raw-moment
// rewrite of the Pearson correlation). Hot loop is fully branchless: all lane
// predicates are combined with bitwise &/| (no short-circuit) so clang emits
// s_and/s_or mask logic + v_cndmask instead of exec-mask branches.
// Block reduction: v_wmma_f32_16x16x4_f32 compresses 32 lanes -> 16 column
// sums per slot (independent WMMAs pipeline), lanes 0..15 store to LDS,
// 26 consumer threads finish with ds_load_b128.
// ---------------------------------------------------------------------------

typedef __attribute__((ext_vector_type(2))) float v2f;
typedef __attribute__((ext_vector_type(4))) float v4f;
typedef __attribute__((ext_vector_type(8))) float v8f;

#define T_DIM  20
#define B_DIM  8
#define HW     230400          // 480*480
#define HW4    57600           // HW / 4 (float4 chunks)
#define NBLK   128             // blocks per time step
#define NTHR   256             // threads per block (8 waves)
#define NWAVE  (NTHR / 32)
#define NSLOT  26              // accumulator slots per (t, block)

// log(30 + 1)
#define LOGF   3.4339872044851463f

// WMMA 32->16 compression: A = ones(16x4), B = {v, 0}. D[m][n] = sum_k B[k][n]
// (column sums, identical in every row). Row M=0 of D = c[0] on lanes 0..15
// (documented C/D layout). Each of the 32 lane inputs lands in exactly one
// column, so the 16 column sums on lanes 0..15 partition the 32-lane total.
__device__ __forceinline__ float wave_colsum(float v) {
  v2f a = {1.0f, 1.0f};
  v2f b = {v, 0.0f};
  v8f c = {0.f, 0.f, 0.f, 0.f, 0.f, 0.f, 0.f, 0.f};
  c = __builtin_amdgcn_wmma_f32_16x16x4_f32(
      /*neg_a=*/false, a, /*neg_b=*/false, b,
      /*c_mod=*/(short)0, c, /*reuse_a=*/false, /*reuse_b=*/false);
  return c[0];
}

// Full 32-lane sum (used only in the tiny per-t kernel): WMMA + 4 xor steps.
__device__ __forceinline__ float wave_sum32(float v) {
  float d = wave_colsum(v);
  d += __shfl_xor(d, 1);
  d += __shfl_xor(d, 2);
  d += __shfl_xor(d, 4);
  d += __shfl_xor(d, 8);
  return d;   // valid on lane 0 (lanes 0..15 in fact)
}

// ---------------------------------------------------------------------------
// Kernel 1: streaming pass. grid = (NBLK, T). Slot layout per (t, block):
//   0..5   : n, Sp, St, Spt, Spp, Stt            (mc-masked raw moments)
//   6..10  : cumulative cnt_t  (valid & q >= thr[l])       -- telescoped later
//   11..15 : cumulative cnt_p  (valid & p >= thr[l])
//   16..20 : cumulative mae    (valid & q >= thr[l]) * |p-q|
//   21..25 : hits per level    (valid & p,q both in level l)
// ---------------------------------------------------------------------------
__global__ __launch_bounds__(NTHR)
void met_partial(const float* __restrict__ pred,
                 const float* __restrict__ tgt,
                 const float* __restrict__ msk,
                 float* __restrict__ ws) {
  const int t   = blockIdx.y;
  const int tid = threadIdx.x;

  float a0 = 0.f, a1 = 0.f, a2 = 0.f, a3 = 0.f, a4 = 0.f, a5 = 0.f;
  float ct[5] = {0.f, 0.f, 0.f, 0.f, 0.f};
  float cp[5] = {0.f, 0.f, 0.f, 0.f, 0.f};
  float hh[5] = {0.f, 0.f, 0.f, 0.f, 0.f};
  float me[5] = {0.f, 0.f, 0.f, 0.f, 0.f};

  for (int b = 0; b < B_DIM; ++b) {
    const size_t base = ((size_t)(b * T_DIM + t)) * (size_t)HW;
    const v4f* p4 = (const v4f*)(pred + base);
    const v4f* t4 = (const v4f*)(tgt + base);
    const v4f* m4 = (const v4f*)(msk + base);

    for (int i = blockIdx.x * NTHR + tid; i < HW4; i += NBLK * NTHR) {
      v4f pv = __builtin_nontemporal_load(p4 + i);
      v4f tv = __builtin_nontemporal_load(t4 + i);
      v4f mv = __builtin_nontemporal_load(m4 + i);
#pragma unroll
      for (int k = 0; k < 4; ++k) {
        float p = fmaxf(__expf(pv[k] * LOGF) - 1.f, 0.f);
        float q = fmaxf(__expf(tv[k] * LOGF) - 1.f, 0.f);
        const bool v = mv[k] > 0.5f;

        // all threshold compares up front -> pure mask logic afterwards
        bool pg[5], qg[5];
        pg[0] = p >= 0.1f; pg[1] = p >= 1.f; pg[2] = p >= 2.f;
        pg[3] = p >= 5.f;  pg[4] = p >= 8.f;
        qg[0] = q >= 0.1f; qg[1] = q >= 1.f; qg[2] = q >= 2.f;
        qg[3] = q >= 5.f;  qg[4] = q >= 8.f;

        // NOTE: bitwise &/| only (no short-circuit -> no exec-mask branches)
        const float mc = (v & (pg[0] | qg[0])) ? 1.f : 0.f;
        a0 += mc;
        a1 = fmaf(p, mc, a1);
        a2 = fmaf(q, mc, a2);
        a3 = fmaf(p * q, mc, a3);
        a4 = fmaf(p * p, mc, a4);
        a5 = fmaf(q * q, mc, a5);

        const float ad = fabsf(p - q);
#pragma unroll
        for (int l = 0; l < 5; ++l) {
          const float ft = (v & qg[l]) ? 1.f : 0.f;    // cumulative cnt_t sel
          ct[l] += ft;
          me[l] = fmaf(ft, ad, me[l]);                 // cumulative mae
          cp[l] += (v & pg[l]) ? 1.f : 0.f;            // cumulative cnt_p
        }
        // exact per-level hits (cannot telescope) — bitwise mask logic only
        bool ip[5], it[5];
        ip[0] = pg[0] & !pg[1]; ip[1] = pg[1] & !pg[2];
        ip[2] = pg[2] & !pg[3]; ip[3] = pg[3] & !pg[4]; ip[4] = pg[4];
        it[0] = qg[0] & !qg[1]; it[1] = qg[1] & !qg[2];
        it[2] = qg[2] & !qg[3]; it[3] = qg[3] & !qg[4]; it[4] = qg[4];
#pragma unroll
        for (int l = 0; l < 5; ++l)
          hh[l] += (v & ip[l] & it[l]) ? 1.f : 0.f;
      }
    }
  }

  float acc[NSLOT];
  acc[0] = a0; acc[1] = a1; acc[2] = a2; acc[3] = a3; acc[4] = a4; acc[5] = a5;
#pragma unroll
  for (int l = 0; l < 5; ++l) {
    acc[6 + l]  = ct[l];
    acc[11 + l] = cp[l];
    acc[16 + l] = me[l];
    acc[21 + l] = hh[l];
  }

  // --- block reduction: WMMA 32->16 per slot, then LDS + b128 sums ---------
  __shared__ float red[NWAVE][NSLOT][16];
  const int wid  = tid >> 5;
  const int lane = tid & 31;
#pragma unroll
  for (int s = 0; s < NSLOT; ++s) {
    const float cs = wave_colsum(acc[s]);   // EXEC all-ones at this point
    if (lane < 16) red[wid][s][lane] = cs;
  }
  __syncthreads();
  if (tid < NSLOT) {
    v4f sum = {0.f, 0.f, 0.f, 0.f};
#pragma unroll
    for (int w = 0; w < NWAVE; ++w) {
      const v4f* rp = (const v4f*)&red[w][tid][0];
#pragma unroll
      for (int j = 0; j < 4; ++j) sum += rp[j];      // ds_load_b128
    }
    ws[((size_t)t * NBLK + blockIdx.x) * NSLOT + tid] =
        sum[0] + sum[1] + sum[2] + sum[3];
  }
}

// ---------------------------------------------------------------------------
// Kernel 2: one wave per time step. Reduce NBLK partials per slot, telescope
// the cumulative slots, compute r_time, ts_time, mae_time, scores_time.
// ---------------------------------------------------------------------------
__global__ __launch_bounds__(32)
void met_final_t(const float* __restrict__ ws, float* __restrict__ out) {
  const int t    = blockIdx.x;
  const int lane = threadIdx.x;

  float S[NSLOT];
#pragma unroll
  for (int s = 0; s < NSLOT; ++s) {
    float v = 0.f;
    for (int b = lane; b < NBLK; b += 32)
      v += ws[((size_t)t * NBLK + b) * NSLOT + s];
    S[s] = wave_sum32(v);                  // valid on lane 0
  }

  if (lane == 0) {
    const float LW[5] = {0.1f, 0.1f, 0.2f, 0.25f, 0.35f};
    float n = S[0];
    float r = 0.f;
    if (n > 0.f) {
      float inv = 1.f / n;
      float num = S[3] - S[1] * S[2] * inv;
      float vp  = fmaxf(S[4] - S[1] * S[1] * inv, 0.f);
      float vt  = fmaxf(S[5] - S[2] * S[2] * inv, 0.f);
      float den = sqrtf(vp * vt);
      r = num / (den + 1e-6f);
      r = fminf(fmaxf(r, -1.f), 1.f);
    }
    float term_corr = sqrtf(expf(r - 1.f));
    float score = 0.f;
#pragma unroll
    for (int l = 0; l < 5; ++l) {
      // telescope cumulative slots back to exact per-level values
      const float c_t = S[6 + l]  - ((l < 4) ? S[7 + l]  : 0.f);
      const float c_p = S[11 + l] - ((l < 4) ? S[12 + l] : 0.f);
      const float ms  = S[16 + l] - ((l < 4) ? S[17 + l] : 0.f);
      const float h   = S[21 + l];
      float ts  = h / (c_t + c_p - h + 1e-8f);   // hits+misses+fas
      float mae = (c_t > 0.f) ? ms / fmaxf(c_t, 1.f) : 0.f;
      out[41 + t * 5 + l]  = ts;
      out[141 + t * 5 + l] = mae;
      score += LW[l] * ts * sqrtf(expf(-mae * 0.01f));
    }
    out[1 + t]  = score * term_corr;   // scores_time
    out[21 + t] = r;                   // r_time
  }
}

// ---------------------------------------------------------------------------
// Kernel 3: scalar epilogue — total_score, ts_levels, mae_levels.
// ---------------------------------------------------------------------------
__global__ void met_final_all(float* __restrict__ out) {
  if (threadIdx.x != 0 || blockIdx.x != 0) return;
  const float TW[T_DIM] = {0.0075f, 0.02f, 0.03f, 0.04f, 0.05f,
                           0.06f,   0.07f, 0.08f, 0.09f, 0.1f,
                           0.09f,   0.08f, 0.07f, 0.06f, 0.05f,
                           0.04f,   0.03f, 0.02f, 0.0075f, 0.005f};
  float tot = 0.f;
  for (int t = 0; t < T_DIM; ++t) tot += out[1 + t] * TW[t];
  out[0] = tot;
  for (int l = 0; l < 5; ++l) {
    float a = 0.f, m = 0.f;
    for (int t = 0; t < T_DIM; ++t) {
      a += out[41 + t * 5 + l];
      m += out[141 + t * 5 + l];
    }
    out[241 + l] = a * (1.f / (float)T_DIM);
    out[246 + l] = m * (1.f / (float)T_DIM);
  }
}

// ---------------------------------------------------------------------------
extern "C" void kernel_launch(void* const* d_in, const int* in_sizes, int n_in,
                              void* d_out, int out_size, void* d_ws, size_t ws_size,
                              hipStream_t stream) {
  const float* pred = (const float*)d_in[0];
  const float* tgt  = (const float*)d_in[1];
  const float* msk  = (const float*)d_in[2];
  float* out = (float*)d_out;
  float* ws  = (float*)d_ws;      // needs T*NBLK*NSLOT*4 = 266240 bytes

  dim3 g1(NBLK, T_DIM);
  met_partial<<<g1, NTHR, 0, stream>>>(pred, tgt, msk, ws);
  met_final_t<<<T_DIM, 32, 0, stream>>>(ws, out);
  met_final_all<<<1, 32, 0, stream>>>(out);
}